// VN_dgcnn_34170759807340
// MI455X (gfx1250) — compile-verified
//
#include <hip/hip_runtime.h>
#include <math.h>

typedef __attribute__((ext_vector_type(2))) float v2f;
typedef __attribute__((ext_vector_type(8))) float v8f;

#define DEVFN __device__ __forceinline__

constexpr int BB   = 4;     // batch
constexpr int NPTS = 1024;  // points
constexpr int KNB  = 20;    // k neighbors
constexpr float EPSV  = 1e-6f;
constexpr float BNEPS = 1e-5f;
constexpr float NSL   = 0.2f;

DEVFN v8f wmma4(v2f a, v2f b, v8f c) {
  // D = A(16x4,f32) * B(4x16,f32) + C(16x16,f32)
  return __builtin_amdgcn_wmma_f32_16x16x4_f32(false, a, false, b, (short)0, c,
                                               false, false);
}

// ---------------------------------------------------------------------------
// Generic WMMA f32 GEMM:  C[M x Ncol] = A * B   (per-batch via blockIdx.z)
// A row-major (lda) or transposed (element = A[k*lda + m]); B row-major KxNcol.
// One wave computes a 16x16 tile. All fragment loads are branch-free:
// addresses are clamped once, OOB contributions zeroed by a 0/1 multiplier.
// Two independent WMMA accumulation chains hide the WMMA->WMMA RAW hazard.
// ---------------------------------------------------------------------------
__global__ void k_wmma_gemm(const float* __restrict__ A, const float* __restrict__ Bm,
                            float* __restrict__ C, int M, int Ncol, int Kd,
                            int lda, int transA, long sA, long sB, long sC) {
  int lane = threadIdx.x & 31;
  int bz = blockIdx.z;
  const float* Ab = A + sA * bz;
  const float* Bb = Bm + sB * bz;
  float* Cb = C + sC * bz;
  int col = blockIdx.x * 16 + (lane & 15);
  int row = blockIdx.y * 16 + (lane & 15);
  int ksel = (lane >> 4) * 2;  // lanes 16-31 hold K+2 per ISA fragment layout
  int row_c = row < M ? row : M - 1;
  int col_c = col < Ncol ? col : Ncol - 1;
  float mA = row < M ? 1.f : 0.f;
  float mB = col < Ncol ? 1.f : 0.f;
  long baseA = transA ? (long)row_c : (long)row_c * lda;
  long strideKA = transA ? (long)lda : 1L;
  const float* pB = Bb + col_c;

  __builtin_prefetch(Ab + baseA, 0, 1);
  __builtin_prefetch(pB, 0, 1);

  v8f acc0 = {}, acc1 = {};
  int KdMain = Kd & ~3;
  int k0 = 0;
  for (; k0 + 8 <= KdMain; k0 += 8) {
    v2f a0, b0, a1, b1;
#pragma unroll
    for (int r = 0; r < 2; ++r) {
      int ka = k0 + ksel + r;
      a0[r] = mA * Ab[baseA + (long)ka * strideKA];
      b0[r] = mB * pB[(long)ka * Ncol];
      a1[r] = mA * Ab[baseA + (long)(ka + 4) * strideKA];
      b1[r] = mB * pB[(long)(ka + 4) * Ncol];
    }
    acc0 = wmma4(a0, b0, acc0);
    acc1 = wmma4(a1, b1, acc1);
  }
  for (; k0 < KdMain; k0 += 4) {
    v2f a0, b0;
#pragma unroll
    for (int r = 0; r < 2; ++r) {
      int ka = k0 + ksel + r;
      a0[r] = mA * Ab[baseA + (long)ka * strideKA];
      b0[r] = mB * pB[(long)ka * Ncol];
    }
    acc0 = wmma4(a0, b0, acc0);
  }
  if (KdMain < Kd) {  // masked tail chunk, branch-free (clamped addresses)
    v2f a0, b0;
#pragma unroll
    for (int r = 0; r < 2; ++r) {
      int ka = KdMain + ksel + r;
      int kac = ka < Kd ? ka : Kd - 1;
      float av = mA * Ab[baseA + (long)kac * strideKA];
      float bv = mB * pB[(long)kac * Ncol];
      a0[r] = ka < Kd ? av : 0.f;
      b0[r] = ka < Kd ? bv : 0.f;
    }
    acc0 = wmma4(a0, b0, acc0);
  }
  v8f acc = acc0 + acc1;

  int rbase = blockIdx.y * 16 + (lane >> 4) * 8;  // C: rows p / p+8 per lane half
#pragma unroll
  for (int p = 0; p < 8; ++p) {
    int rr = rbase + p;
    if (rr < M && col < Ncol) Cb[(long)rr * Ncol + col] = acc[p];
  }
}

// ---------------------------------------------------------------------------
// Graph-feature GEMM: P[b][o][axis][n][kk] = sum_kc W[o][kc] * feat(kc,axis,n,kk)
// feat computed on the fly from x (B,C,3,N) and knn idx (B,N,K):
//   kc <  C : x[c][axis][nbr] - x[c][axis][n]
//   kc >= C : x[c-C][axis][n]
// Never materializes the (B,2C,3,N,K) tensor. Cin2 = 2C (2, 42 or 84 here) is not always a multiple
// of 4, yet the K loop has no masked tail; all selects are branch-free cndmasks.
// ---------------------------------------------------------------------------
__global__ void k_graph_gemm(const float* __restrict__ W, const float* __restrict__ X,
                             const int* __restrict__ idx, float* __restrict__ P,
                             int Cout, int Cin2, int C) {
  constexpr int M3 = 3 * NPTS * KNB;
  int lane = threadIdx.x & 31;
  int bz = blockIdx.z;
  int col = blockIdx.x * 16 + (lane & 15);
  int rowv = blockIdx.y * 16 + (lane & 15);
  int ksel = (lane >> 4) * 2;
  int col_c = col < M3 ? col : M3 - 1;
  float mB = col < M3 ? 1.f : 0.f;
  int row_c = rowv < Cout ? rowv : Cout - 1;
  float mA = rowv < Cout ? 1.f : 0.f;

  int axis = col_c / (NPTS * KNB);
  int rem = col_c - axis * (NPTS * KNB);
  int n = rem / KNB;
  int kkk = rem - n * KNB;
  int nbr = idx[((long)bz * NPTS + n) * KNB + kkk];
  const float* xb = X + (long)bz * C * 3 * NPTS + (long)axis * NPTS;
  const float* wr = W + (long)row_c * Cin2;
  __builtin_prefetch(wr, 0, 1);

  v8f acc0 = {}, acc1 = {};
  int k0 = 0;
  for (; k0 + 8 <= Cin2; k0 += 8) {
    v2f a0, b0, a1, b1;
#pragma unroll
    for (int r = 0; r < 2; ++r) {
#pragma unroll
      for (int u = 0; u < 2; ++u) {
        int ka = k0 + ksel + r + 4 * u;
        int c = ka < C ? ka : ka - C;
        const float* xc = xb + (long)c * 3 * NPTS;
        float ctr = xc[n];
        float nv = xc[nbr];
        float bv = mB * (ka < C ? (nv - ctr) : ctr);
        float av = mA * wr[ka];
        if (u == 0) { a0[r] = av; b0[r] = bv; }
        else        { a1[r] = av; b1[r] = bv; }
      }
    }
    acc0 = wmma4(a0, b0, acc0);
    acc1 = wmma4(a1, b1, acc1);
  }
  for (; k0 < Cin2; k0 += 4) {
    v2f a0, b0;
#pragma unroll
    for (int r = 0; r < 2; ++r) {
      int ka = k0 + ksel + r;
      int c = ka < C ? ka : ka - C;
      const float* xc = xb + (long)c * 3 * NPTS;
      float ctr = xc[n];
      float nv = xc[nbr];
      b0[r] = mB * (ka < C ? (nv - ctr) : ctr);
      a0[r] = mA * wr[ka];
    }
    acc0 = wmma4(a0, b0, acc0);
  }
  v8f acc = acc0 + acc1;

  float* Pb = P + (long)bz * Cout * M3;
  int rbase = blockIdx.y * 16 + (lane >> 4) * 8;
#pragma unroll
  for (int p = 0; p < 8; ++p) {
    int rr = rbase + p;
    if (rr < Cout && col < M3) Pb[(long)rr * M3 + col] = acc[p];
  }
}

// ---------------------------------------------------------------------------
__global__ void k_transpose_in(const float* __restrict__ x, float* __restrict__ x0) {
  int t = blockIdx.x * blockDim.x + threadIdx.x;  // (b, axis, n)
  if (t >= BB * 3 * NPTS) return;
  int b = t / (3 * NPTS);
  int r = t - b * 3 * NPTS;
  int a = r / NPTS;
  int n = r - a * NPTS;
  x0[t] = x[((long)b * NPTS + n) * 3 + a];
}

__global__ void k_sqnorm(const float* __restrict__ X, float* __restrict__ xx, int D) {
  int t = blockIdx.x * blockDim.x + threadIdx.x;
  if (t >= BB * NPTS) return;
  int b = t / NPTS, n = t - b * NPTS;
  const float* xb = X + (long)b * D * NPTS;
  float s = 0.f;
  for (int d = 0; d < D; ++d) { float v = xb[(long)d * NPTS + n]; s += v * v; }
  xx[t] = s;
}

// one wave per row: dist[j] = xx[i]+xx[j]-2*S[i][j]; 20 iterative min-extracts
__global__ void k_topk(const float* __restrict__ S, const float* __restrict__ xx,
                       int* __restrict__ idx) {
  __shared__ float dsh[NPTS];
  int b = blockIdx.y, i = blockIdx.x, lane = threadIdx.x;
  float xi = xx[b * NPTS + i];
  const float* Srow = S + ((long)b * NPTS + i) * NPTS;
  for (int j = lane; j < NPTS; j += 32)
    dsh[j] = xi + xx[b * NPTS + j] - 2.f * Srow[j];
  __syncthreads();
  for (int t = 0; t < KNB; ++t) {
    float best = 1e30f; int bj = NPTS;
    for (int j = lane; j < NPTS; j += 32) {
      float v = dsh[j];
      if (v < best || (v == best && j < bj)) { best = v; bj = j; }
    }
#pragma unroll
    for (int off = 16; off > 0; off >>= 1) {
      float ov = __shfl_xor(best, off, 32);
      int   oj = __shfl_xor(bj, off, 32);
      if (ov < best || (ov == best && oj < bj)) { best = ov; bj = oj; }
    }
    if (lane == 0) {
      idx[((long)b * NPTS + i) * KNB + t] = bj;
      dsh[bj] = 1e30f;
    }
    __syncthreads();
  }
}

// per-channel sum / sumsq of vector norms, LDS tree reduce + 2 atomics/block
__global__ void k_norm_stats(const float* __restrict__ P, float* __restrict__ stats,
                             int Cout, int NK) {
  int o = blockIdx.x, b = blockIdx.y;
  const float* pb = P + ((long)b * Cout + o) * 3L * NK;
  float s = 0.f, ss = 0.f;
  for (int m = threadIdx.x; m < NK; m += blockDim.x) {
    float a0 = pb[m], a1 = pb[NK + m], a2 = pb[2L * NK + m];
    float nn = sqrtf(a0 * a0 + a1 * a1 + a2 * a2) + EPSV;
    s += nn; ss += nn * nn;
  }
  __shared__ float sh0[256], sh1[256];
  int tid = threadIdx.x;
  sh0[tid] = s; sh1[tid] = ss;
  __syncthreads();
  for (int w = 128; w > 0; w >>= 1) {
    if (tid < w) { sh0[tid] += sh0[tid + w]; sh1[tid] += sh1[tid + w]; }
    __syncthreads();
  }
  if (tid == 0) {
    atomicAdd(&stats[2 * o], sh0[0]);
    atomicAdd(&stats[2 * o + 1], sh1[0]);
  }
}

__global__ void k_finalize_stats(float* __restrict__ stats, int Cout, float invcount) {
  int o = blockIdx.x * blockDim.x + threadIdx.x;
  if (o >= Cout) return;
  float m = stats[2 * o] * invcount;
  float v = stats[2 * o + 1] * invcount - m * m;
  stats[2 * o] = m;
  stats[2 * o + 1] = rsqrtf(v + BNEPS);
}

// VN batchnorm + directional leaky-relu + mean-pool over k
__global__ void k_vn_apply(const float* __restrict__ P, const float* __restrict__ D,
                           const float* __restrict__ stats, const float* __restrict__ gpar,
                           const float* __restrict__ bpar, float* __restrict__ out,
                           int Cout, int Nsp, int Kpool, int share) {
  long t = (long)blockIdx.x * blockDim.x + threadIdx.x;
  long total = (long)BB * Cout * Nsp;
  if (t >= total) return;
  int n = (int)(t % Nsp);
  long t2 = t / Nsp;
  int o = (int)(t2 % Cout);
  int b = (int)(t2 / Cout);
  long NKl = (long)Nsp * Kpool;
  const float* pb = P + ((long)b * Cout + o) * 3 * NKl;
  int Cd = share ? 1 : Cout;
  int od = share ? 0 : o;
  const float* db = D + ((long)b * Cd + od) * 3 * NKl;
  float mean = stats[2 * o], istd = stats[2 * o + 1];
  float gg = gpar[o], bo = bpar[o];
  float a0 = 0.f, a1 = 0.f, a2 = 0.f;
  for (int kk = 0; kk < Kpool; ++kk) {
    long m = (long)n * Kpool + kk;
    float p0 = pb[m], p1 = pb[NKl + m], p2 = pb[2 * NKl + m];
    float nn = sqrtf(p0 * p0 + p1 * p1 + p2 * p2) + EPSV;
    float nbv = (nn - mean) * istd * gg + bo;
    float sc = nbv / nn;
    p0 *= sc; p1 *= sc; p2 *= sc;
    float d0 = db[m], d1 = db[NKl + m], d2 = db[2 * NKl + m];
    float dot = p0 * d0 + p1 * d1 + p2 * d2;
    float q0 = p0, q1 = p1, q2 = p2;
    if (dot < 0.f) {
      float f = dot / (d0 * d0 + d1 * d1 + d2 * d2 + EPSV);
      q0 -= f * d0; q1 -= f * d1; q2 -= f * d2;
    }
    a0 += NSL * p0 + (1.f - NSL) * q0;
    a1 += NSL * p1 + (1.f - NSL) * q1;
    a2 += NSL * p2 + (1.f - NSL) * q2;
  }
  float inv = 1.f / (float)Kpool;
  float* ob = out + ((long)b * Cout + o) * 3L * Nsp + n;
  ob[0] = a0 * inv; ob[Nsp] = a1 * inv; ob[2L * Nsp] = a2 * inv;
}

__global__ void k_concat4(const float* __restrict__ x1, const float* __restrict__ x2,
                          const float* __restrict__ x3, const float* __restrict__ x4,
                          float* __restrict__ out) {
  long t = (long)blockIdx.x * blockDim.x + threadIdx.x;
  const long per = 169L * 3 * NPTS;
  if (t >= BB * per) return;
  int b = (int)(t / per);
  long r = t - (long)b * per;
  int c = (int)(r / (3 * NPTS));
  long r2 = r - (long)c * 3 * NPTS;
  float v;
  if (c < 21)       v = x1[((long)b * 21 + c) * 3 * NPTS + r2];
  else if (c < 42)  v = x2[((long)b * 21 + (c - 21)) * 3 * NPTS + r2];
  else if (c < 84)  v = x3[((long)b * 42 + (c - 42)) * 3 * NPTS + r2];
  else              v = x4[((long)b * 85 + (c - 84)) * 3 * NPTS + r2];
  out[t] = v;
}

__global__ void k_mean_n(const float* __restrict__ h, float* __restrict__ hm, int C) {
  int t = blockIdx.x * blockDim.x + threadIdx.x;  // (b, c, axis)
  if (t >= BB * C * 3) return;
  const float* r = h + (long)t * NPTS;
  float s = 0.f;
  for (int n = 0; n < NPTS; ++n) s += r[n];
  hm[t] = s * (1.f / NPTS);
}

__global__ void k_build_hc(const float* __restrict__ h5, const float* __restrict__ hm,
                           float* __restrict__ hc) {
  long t = (long)blockIdx.x * blockDim.x + threadIdx.x;
  const long per = 682L * 3 * NPTS;
  if (t >= BB * per) return;
  int b = (int)(t / per);
  long r = t - (long)b * per;
  int c = (int)(r / (3 * NPTS));
  long r2 = r - (long)c * 3 * NPTS;
  if (c < 341) hc[t] = h5[((long)b * 341 + c) * 3 * NPTS + r2];
  else         hc[t] = hm[((long)b * 341 + (c - 341)) * 3 + (int)(r2 / NPTS)];
}

// hs[b][i][j2][n] = sum_j hc[b][i][j][n] * z0[b][j][j2][n]
__global__ void k_hs(const float* __restrict__ hc, const float* __restrict__ z0,
                     float* __restrict__ hs) {
  long t = (long)blockIdx.x * blockDim.x + threadIdx.x;
  long total = (long)BB * 682 * 3 * NPTS;
  if (t >= total) return;
  int n = (int)(t % NPTS);
  long t2 = t / NPTS;
  int j2 = (int)(t2 % 3);
  long t3 = t2 / 3;
  int i = (int)(t3 % 682);
  int b = (int)(t3 / 682);
  float s = 0.f;
#pragma unroll
  for (int j = 0; j < 3; ++j)
    s += hc[(((long)b * 682 + i) * 3 + j) * NPTS + n] *
         z0[(((long)b * 3 + j) * 3 + j2) * NPTS + n];
  hs[t] = s;
}

__global__ void k_pool(const float* __restrict__ hs, float* __restrict__ feats) {
  int c = blockIdx.x, b = blockIdx.y;
  const float* row = hs + ((long)b * 2046 + c) * NPTS;
  float mx = -1e30f, sm = 0.f;
  for (int n = threadIdx.x; n < NPTS; n += 256) {
    float v = row[n];
    mx = fmaxf(mx, v); sm += v;
  }
  __shared__ float shm[256], shs[256];
  int tid = threadIdx.x;
  shm[tid] = mx; shs[tid] = sm;
  __syncthreads();
  for (int w = 128; w > 0; w >>= 1) {
    if (tid < w) { shm[tid] = fmaxf(shm[tid], shm[tid + w]); shs[tid] += shs[tid + w]; }
    __syncthreads();
  }
  if (tid == 0) {
    feats[(long)b * 4092 + c] = shm[0];
    feats[(long)b * 4092 + 2046 + c] = shs[0] * (1.f / NPTS);
  }
}

__global__ void k_linear(const float* __restrict__ W, const float* __restrict__ bias,
                         const float* __restrict__ in, float* __restrict__ out,
                         int O, int I) {
  int t = blockIdx.x * blockDim.x + threadIdx.x;
  if (t >= BB * O) return;
  int b = t / O, o = t - b * O;
  const float* wr = W + (long)o * I;
  const float* ir = in + (long)b * I;
  float s = bias[o];
  for (int i = 0; i < I; ++i) s += wr[i] * ir[i];
  out[(long)b * O + o] = s;
}

__global__ void k_bn1d_leaky(float* __restrict__ h, const float* __restrict__ gpar,
                             const float* __restrict__ be, int O) {
  int o = blockIdx.x * blockDim.x + threadIdx.x;
  if (o >= O) return;
  float m = 0.f;
  for (int b = 0; b < BB; ++b) m += h[(long)b * O + o];
  m *= (1.f / BB);
  float v = 0.f;
  for (int b = 0; b < BB; ++b) { float d = h[(long)b * O + o] - m; v += d * d; }
  v *= (1.f / BB);
  float istd = rsqrtf(v + BNEPS);
  float gg = gpar[o], bb = be[o];
  for (int b = 0; b < BB; ++b) {
    float val = (h[(long)b * O + o] - m) * istd * gg + bb;
    h[(long)b * O + o] = val >= 0.f ? val : NSL * val;
  }
}

// ---------------------------------------------------------------------------
extern "C" void kernel_launch(void* const* d_in, const int* in_sizes, int n_in,
                              void* d_out, int out_size, void* d_ws, size_t ws_size,
                              hipStream_t stream) {
  (void)in_sizes; (void)n_in; (void)out_size; (void)ws_size;
  const float* X = (const float*)d_in[0];
  const float* Wf[7]; const float* Wd[7]; const float* Gp[7]; const float* Bp[7];
  for (int l = 0; l < 7; ++l) {
    Wf[l] = (const float*)d_in[1 + 4 * l];
    Wd[l] = (const float*)d_in[2 + 4 * l];
    Gp[l] = (const float*)d_in[3 + 4 * l];
    Bp[l] = (const float*)d_in[4 + 4 * l];
  }
  const float* vnlin = (const float*)d_in[29];
  const float* W1 = (const float*)d_in[30]; const float* b1 = (const float*)d_in[31];
  const float* g1 = (const float*)d_in[32]; const float* be1 = (const float*)d_in[33];
  const float* W2 = (const float*)d_in[34]; const float* b2 = (const float*)d_in[35];
  const float* g2 = (const float*)d_in[36]; const float* be2 = (const float*)d_in[37];
  const float* W3 = (const float*)d_in[38]; const float* b3 = (const float*)d_in[39];

  char* cur = (char*)d_ws;
  auto allocb = [&](size_t bytes) { char* p = cur; cur += (bytes + 255) & ~(size_t)255; return p; };
  auto allocf = [&](size_t n) { return (float*)allocb(n * sizeof(float)); };

  float* x0   = allocf((size_t)BB * 3 * NPTS);
  float* xxb  = allocf((size_t)BB * NPTS);
  float* stats = allocf(1024);
  int*   idxb = (int*)allocb((size_t)BB * NPTS * KNB * sizeof(int));
  float* x1   = allocf((size_t)BB * 21 * 3 * NPTS);
  float* x2   = allocf((size_t)BB * 21 * 3 * NPTS);
  float* x3   = allocf((size_t)BB * 42 * 3 * NPTS);
  float* x4   = allocf((size_t)BB * 85 * 3 * NPTS);
  float* hcat = allocf((size_t)BB * 169 * 3 * NPTS);
  float* h5   = allocf((size_t)BB * 341 * 3 * NPTS);
  float* hmb  = allocf((size_t)BB * 341 * 3);
  float* hcb  = allocf((size_t)BB * 682 * 3 * NPTS);
  float* z1   = allocf((size_t)BB * 341 * 3 * NPTS);
  float* z2   = allocf((size_t)BB * 170 * 3 * NPTS);
  float* z0b  = allocf((size_t)BB * 3 * 3 * NPTS);
  float* feats = allocf((size_t)BB * 4092);
  float* h1   = allocf((size_t)BB * 512);
  float* h2   = allocf((size_t)BB * 256);
  float* pbuf = allocf((size_t)BB * 85 * 3 * NPTS * KNB);  // largest p (conv4)
  float* dbuf = allocf((size_t)BB * 85 * 3 * NPTS * KNB);
  float* Sbuf = dbuf;   // Gram matrix overlays d (free at that point)
  float* hsb  = pbuf;   // hs overlays p (free after vn2)

  auto gemm = [&](const float* A, const float* Bm, float* C, int M, int Ncol, int Kd,
                  int lda, int transA, long sA, long sB, long sC) {
    dim3 g((Ncol + 15) / 16, (M + 15) / 16, BB);
    k_wmma_gemm<<<g, 32, 0, stream>>>(A, Bm, C, M, Ncol, Kd, lda, transA, sA, sB, sC);
  };

  auto graph_layer = [&](const float* xin, int C, int Cout, const float* wf,
                         const float* wd, const float* gp, const float* bp, float* xout) {
    int D = 3 * C;
    k_sqnorm<<<(BB * NPTS + 255) / 256, 256, 0, stream>>>(xin, xxb, D);
    // Gram S = xin^T * xin  (A transposed, lda = NPTS)
    gemm(xin, xin, Sbuf, NPTS, NPTS, D, NPTS, 1,
         (long)D * NPTS, (long)D * NPTS, (long)NPTS * NPTS);
    k_topk<<<dim3(NPTS, BB), 32, 0, stream>>>(Sbuf, xxb, idxb);
    const int M3 = 3 * NPTS * KNB;
    dim3 gg((M3 + 15) / 16, (Cout + 15) / 16, BB);
    k_graph_gemm<<<gg, 32, 0, stream>>>(wf, xin, idxb, pbuf, Cout, 2 * C, C);
    k_graph_gemm<<<gg, 32, 0, stream>>>(wd, xin, idxb, dbuf, Cout, 2 * C, C);
    hipMemsetAsync(stats, 0, (size_t)Cout * 2 * sizeof(float), stream);
    k_norm_stats<<<dim3(Cout, BB), 256, 0, stream>>>(pbuf, stats, Cout, NPTS * KNB);
    k_finalize_stats<<<(Cout + 63) / 64, 64, 0, stream>>>(
        stats, Cout, 1.f / ((float)BB * NPTS * KNB));
    long tot = (long)BB * Cout * NPTS;
    k_vn_apply<<<(tot + 255) / 256, 256, 0, stream>>>(pbuf, dbuf, stats, gp, bp, xout,
                                                      Cout, NPTS, KNB, 0);
  };

  auto vn_layer = [&](const float* xin, int Cin, int Cout, const float* wf,
                      const float* wd, const float* gp, const float* bp, float* xout,
                      int share) {
    int Ncol = 3 * NPTS;
    gemm(wf, xin, pbuf, Cout, Ncol, Cin, Cin, 0, 0,
         (long)Cin * Ncol, (long)Cout * Ncol);
    int Cd = share ? 1 : Cout;
    gemm(wd, xin, dbuf, Cd, Ncol, Cin, Cin, 0, 0,
         (long)Cin * Ncol, (long)Cd * Ncol);
    hipMemsetAsync(stats, 0, (size_t)Cout * 2 * sizeof(float), stream);
    k_norm_stats<<<dim3(Cout, BB), 256, 0, stream>>>(pbuf, stats, Cout, NPTS);
    k_finalize_stats<<<(Cout + 63) / 64, 64, 0, stream>>>(
        stats, Cout, 1.f / ((float)BB * NPTS));
    long tot = (long)BB * Cout * NPTS;
    k_vn_apply<<<(tot + 255) / 256, 256, 0, stream>>>(pbuf, dbuf, stats, gp, bp, xout,
                                                      Cout, NPTS, 1, share);
  };

  // ---- pipeline ----
  k_transpose_in<<<(BB * 3 * NPTS + 255) / 256, 256, 0, stream>>>(X, x0);
  graph_layer(x0, 1, 21, Wf[0], Wd[0], Gp[0], Bp[0], x1);
  graph_layer(x1, 21, 21, Wf[1], Wd[1], Gp[1], Bp[1], x2);
  graph_layer(x2, 21, 42, Wf[2], Wd[2], Gp[2], Bp[2], x3);
  graph_layer(x3, 42, 85, Wf[3], Wd[3], Gp[3], Bp[3], x4);

  {
    long tot = (long)BB * 169 * 3 * NPTS;
    k_concat4<<<(tot + 255) / 256, 256, 0, stream>>>(x1, x2, x3, x4, hcat);
  }
  vn_layer(hcat, 169, 341, Wf[4], Wd[4], Gp[4], Bp[4], h5, 1);   // conv5 (shared Wd)
  k_mean_n<<<(BB * 341 * 3 + 255) / 256, 256, 0, stream>>>(h5, hmb, 341);
  {
    long tot = (long)BB * 682 * 3 * NPTS;
    k_build_hc<<<(tot + 255) / 256, 256, 0, stream>>>(h5, hmb, hcb);
  }
  vn_layer(hcb, 682, 341, Wf[5], Wd[5], Gp[5], Bp[5], z1, 0);    // vn1
  vn_layer(z1, 341, 170, Wf[6], Wd[6], Gp[6], Bp[6], z2, 0);     // vn2
  // z0 = vn_lin (3x170) * z2
  gemm(vnlin, z2, z0b, 3, 3 * NPTS, 170, 170, 0, 0,
       (long)170 * 3 * NPTS, (long)3 * 3 * NPTS);
  {
    long tot = (long)BB * 682 * 3 * NPTS;
    k_hs<<<(tot + 255) / 256, 256, 0, stream>>>(hcb, z0b, hsb);
  }
  k_pool<<<dim3(2046, BB), 256, 0, stream>>>(hsb, feats);
  k_linear<<<(BB * 512 + 255) / 256, 256, 0, stream>>>(W1, b1, feats, h1, 512, 4092);
  k_bn1d_leaky<<<(512 + 63) / 64, 64, 0, stream>>>(h1, g1, be1, 512);
  k_linear<<<(BB * 256 + 255) / 256, 256, 0, stream>>>(W2, b2, h1, h2, 256, 512);
  k_bn1d_leaky<<<(256 + 63) / 64, 64, 0, stream>>>(h2, g2, be2, 256);
  k_linear<<<(BB * 40 + 255) / 256, 256, 0, stream>>>(W3, b3, h2, (float*)d_out, 40, 256);
}